// GNN_base_5153960755959
// MI455X (gfx1250) — compile-verified
//
#include <hip/hip_runtime.h>
#include <hip/hip_bf16.h>

#define N_NODES 100000
#define N_EDGES 1600000
#define D 128

typedef float v2f __attribute__((ext_vector_type(2)));
typedef float v8f __attribute__((ext_vector_type(8)));

// ---------------------------------------------------------------------------
// Scatter: one wave (32 lanes) per edge. Lane l moves 4 consecutive floats
// (float4 gather from x[src], 4 hardware float atomics into msg[dst]).
// Edge indices are wave-uniform: readfirstlane lets the compiler issue them
// as scalar (s_load) fetches instead of 32-lane vector loads.
// COUNT=true additionally accumulates the per-destination edge count (lane 0).
// ---------------------------------------------------------------------------
template <bool COUNT>
__global__ __launch_bounds__(256) void sage_scatter_kernel(
    const float* __restrict__ x,
    const long long* __restrict__ ei,   // [2, N_EDGES] int64
    float* __restrict__ msg,            // [N_NODES, D] (pre-zeroed)
    float* __restrict__ cnt)            // [N_NODES]    (pre-zeroed)
{
    const unsigned tid  = blockIdx.x * blockDim.x + threadIdx.x;
    const unsigned lane = tid & 31u;
    // e is uniform across the wave; make it provably scalar.
    const int e = __builtin_amdgcn_readfirstlane((int)(tid >> 5));
    if (e >= N_EDGES) return;

    const int src = (int)ei[e];
    const int dst = (int)ei[(size_t)N_EDGES + e];

    const float4 v = *(const float4*)(x + (size_t)src * D + lane * 4);
    float* mp = msg + (size_t)dst * D + lane * 4;
    __hip_atomic_fetch_add(mp + 0, v.x, __ATOMIC_RELAXED, __HIP_MEMORY_SCOPE_AGENT);
    __hip_atomic_fetch_add(mp + 1, v.y, __ATOMIC_RELAXED, __HIP_MEMORY_SCOPE_AGENT);
    __hip_atomic_fetch_add(mp + 2, v.z, __ATOMIC_RELAXED, __HIP_MEMORY_SCOPE_AGENT);
    __hip_atomic_fetch_add(mp + 3, v.w, __ATOMIC_RELAXED, __HIP_MEMORY_SCOPE_AGENT);

    if (COUNT && lane == 0)
        __hip_atomic_fetch_add(cnt + dst, 1.0f, __ATOMIC_RELAXED, __HIP_MEMORY_SCOPE_AGENT);
}

// ---------------------------------------------------------------------------
// Fused SAGE linear: out = relu?( (msg .* inv_cnt) @ Wl + bl + xin @ Wr )
//
// Wave-level tiling with V_WMMA_F32_16X16X4_F32 (exact fp32):
//   block = 256 threads = 8 waves; wave w owns output columns [16w, 16w+16).
//   Each block iterates ROW_ITERS row-blocks of 16 rows. 625 blocks * 10 * 16
//   rows = 100000 exactly -> no divergence, EXEC all-ones (WMMA requirement).
//
// fp32 WMMA fragment layouts (ISA 7.12.2):
//   A (16x4):  lane m = lane%16, VGPR r holds K = 2*(lane/16) + r  (contiguous pair)
//   B (4x16):  lane n = lane%16, VGPR r holds K = kb + r + 2*(lane/16)
//   C/D:       VGPR j holds row  j + 8*(lane/16), col = lane%16
//
// B fragments of Wl and Wr are preloaded once per wave into registers
// (2 matrices * 32 k-steps * v2f = 128 VGPRs) and reused for all row blocks.
// ---------------------------------------------------------------------------
template <bool RELU, int ROW_ITERS>
__global__ __launch_bounds__(256, 1) void sage_gemm_kernel(
    const float* __restrict__ xin,  // [N, D] self features
    const float* __restrict__ msg,  // [N, D] summed messages
    const float* __restrict__ cnt,  // [N]
    const float* __restrict__ Wl,   // [D, D] row-major (k, n)
    const float* __restrict__ bl,   // [D]
    const float* __restrict__ Wr,   // [D, D]
    float* __restrict__ out)        // [N, D]
{
    const int lane = threadIdx.x & 31;
    const int wave = threadIdx.x >> 5;          // 0..7 -> column tile
    const int half = lane >> 4;                 // 0 or 1
    const int n0   = wave * 16;
    const int nc   = n0 + (lane & 15);          // this lane's output column
    const int koff = half * 2;                  // K sub-offset inside a K=4 step

    // ---- preload B fragments for both weight matrices into registers ----
    v2f BL[32], BR[32];
#pragma unroll
    for (int t = 0; t < 32; ++t) {
        const int kb = t * 4 + koff;
        BL[t][0] = Wl[(size_t)(kb + 0) * D + nc];
        BL[t][1] = Wl[(size_t)(kb + 1) * D + nc];
        BR[t][0] = Wr[(size_t)(kb + 0) * D + nc];
        BR[t][1] = Wr[(size_t)(kb + 1) * D + nc];
    }
    const float bias = bl[nc];

    for (int it = 0; it < ROW_ITERS; ++it) {
        const int rb  = blockIdx.x * ROW_ITERS + it;   // row block index
        const int row = rb * 16 + (lane & 15);         // this lane's A row

        // mean divisor: v_rcp_f32 (~1 ULP) instead of IEEE divide sequence
        const float inv = __builtin_amdgcn_rcpf(fmaxf(cnt[row], 1.0f));

        v8f acc;
#pragma unroll
        for (int j = 0; j < 8; ++j) acc[j] = bias;     // bias splat init

        const float* aggRow  = msg + (size_t)row * D;
        const float* selfRow = xin + (size_t)row * D;

#pragma unroll
        for (int t = 0; t < 32; ++t) {                 // agg @ Wl
            const int kb = t * 4 + koff;
            v2f a;
            a[0] = aggRow[kb + 0] * inv;
            a[1] = aggRow[kb + 1] * inv;
            acc = __builtin_amdgcn_wmma_f32_16x16x4_f32(
                false, a, false, BL[t], (short)0, acc, false, false);
        }
#pragma unroll
        for (int t = 0; t < 32; ++t) {                 // x @ Wr
            const int kb = t * 4 + koff;
            v2f a;
            a[0] = selfRow[kb + 0];
            a[1] = selfRow[kb + 1];
            acc = __builtin_amdgcn_wmma_f32_16x16x4_f32(
                false, a, false, BR[t], (short)0, acc, false, false);
        }

        // ---- store D tile: VGPR j -> row rb*16 + j + 8*half, column nc ----
        const int mrow = rb * 16 + half * 8;
#pragma unroll
        for (int j = 0; j < 8; ++j) {
            float v = acc[j];
            if (RELU) v = fmaxf(v, 0.0f);
            out[(size_t)(mrow + j) * D + nc] = v;
        }
    }
}

// ---------------------------------------------------------------------------
extern "C" void kernel_launch(void* const* d_in, const int* in_sizes, int n_in,
                              void* d_out, int out_size, void* d_ws, size_t ws_size,
                              hipStream_t stream) {
    (void)in_sizes; (void)n_in; (void)out_size; (void)ws_size;

    const float*     x   = (const float*)d_in[0];
    const long long* ei  = (const long long*)d_in[1];   // int64 edge_index
    const float*     Wl1 = (const float*)d_in[2];
    const float*     bl1 = (const float*)d_in[3];
    const float*     Wr1 = (const float*)d_in[4];
    const float*     Wl2 = (const float*)d_in[5];
    const float*     bl2 = (const float*)d_in[6];
    const float*     Wr2 = (const float*)d_in[7];
    float*           out = (float*)d_out;

    // workspace layout: h [N*D] | msg [N*D] | cnt [N]   (~103 MB)
    float* h   = (float*)d_ws;
    float* msg = h   + (size_t)N_NODES * D;
    float* cnt = msg + (size_t)N_NODES * D;

    const dim3 sblk(256);
    const dim3 sgrid((unsigned)(((size_t)N_EDGES * 32 + 255) / 256));  // 200000
    const dim3 gblk(256);
    const dim3 ggrid(625);  // 625 blocks * 10 iters * 16 rows = 100000

    // ---- layer 1 ----
    hipMemsetAsync(msg, 0, ((size_t)N_NODES * D + N_NODES) * sizeof(float), stream);
    sage_scatter_kernel<true><<<sgrid, sblk, 0, stream>>>(x, ei, msg, cnt);
    sage_gemm_kernel<true, 10><<<ggrid, gblk, 0, stream>>>(x, msg, cnt, Wl1, bl1, Wr1, h);

    // ---- layer 2 (cnt is reused; only msg is re-zeroed) ----
    hipMemsetAsync(msg, 0, (size_t)N_NODES * D * sizeof(float), stream);
    sage_scatter_kernel<false><<<sgrid, sblk, 0, stream>>>(h, ei, msg, cnt);
    sage_gemm_kernel<false, 10><<<ggrid, gblk, 0, stream>>>(h, msg, cnt, Wl2, bl2, Wr2, out);
}